// SpatialGraphConv_77841987273129
// MI455X (gfx1250) — compile-verified
//
#include <hip/hip_runtime.h>
#include <hip/hip_bf16.h>

typedef float v2f __attribute__((ext_vector_type(2)));
typedef float v8f __attribute__((ext_vector_type(8)));

#define DIM 64   // node feature dim = KI + KM
#define HID 64   // mlp hidden
#define KI  32
#define KM  32

__device__ __forceinline__ v8f wmma_f32_16x16x4(v2f a, v2f b, v8f c) {
    // 8 args: (neg_a, A, neg_b, B, c_mod, C, reuse_a, reuse_b)
    return __builtin_amdgcn_wmma_f32_16x16x4_f32(false, a, false, b, (short)0, c,
                                                 false, false);
}

// ---------------------------------------------------------------------------
// Zero the workspace (sums + gathered)
// ---------------------------------------------------------------------------
__global__ void zero_ws_kernel(float* p, int n) {
    int i = blockIdx.x * blockDim.x + threadIdx.x;
    if (i < n) p[i] = 0.0f;
}

__device__ __forceinline__ int indicator_bin(float d) {
    // bins = linspace(0,1,33); indicator=1 iff bins[k] < d < bins[k+1] strictly
    float s = d * (float)KI;
    int   b = (int)floorf(s);
    if (b < 0 || b >= KI) return -1;
    if (!(s > (float)b) || !(s < (float)(b + 1))) return -1;
    return b;
}

// ---------------------------------------------------------------------------
// Edge MLP tile via WMMA f32 16x16x4, B fed from LDS:
//   mlp[16 edges, 32] = relu( relu(d*w1+b1)[16,64] @ w2[64,32] + b2 )
// A fragment (activations) computed once per lane, reused for both col tiles.
// C/D layout: M=(lane>>4)*8+r, N=lane&15.
// ---------------------------------------------------------------------------
__device__ __forceinline__ void mlp_tile_wmma(int tileBase, int Ne,
                                              const float* __restrict__ dist,
                                              const float* sW1, const float* sB1,
                                              const float* sW2, const float* sB2,
                                              int lane, v8f acc[2]) {
    int row   = lane & 15;                    // A-matrix row (edge within tile)
    int e     = min(tileBase + row, Ne - 1);
    float d   = dist[e];
    int khalf = (lane >> 4) * 2;              // K sub-offset per half-wave
    int coln  = lane & 15;                    // B/C column within tile

    // hoisted A fragments: h = relu(d*w1+b1) for this lane's K slots
    v2f hv[16];
#pragma unroll
    for (int ks = 0; ks < 16; ++ks) {
        int jk = ks * 4 + khalf;
        hv[ks].x = fmaxf(fmaf(d, sW1[jk],     sB1[jk]),     0.0f);
        hv[ks].y = fmaxf(fmaf(d, sW1[jk + 1], sB1[jk + 1]), 0.0f);
    }

#pragma unroll
    for (int c = 0; c < 2; ++c) {
        float bias = sB2[c * 16 + coln];
        v8f a_acc;
#pragma unroll
        for (int r = 0; r < 8; ++r) a_acc[r] = bias;
#pragma unroll
        for (int ks = 0; ks < 16; ++ks) {
            int jk = ks * 4 + khalf;
            v2f bv;
            bv.x = sW2[jk       * KM + c * 16 + coln];
            bv.y = sW2[(jk + 1) * KM + c * 16 + coln];
            a_acc = wmma_f32_16x16x4(hv[ks], bv, a_acc);
        }
        acc[c] = a_acc;
    }
}

// ---------------------------------------------------------------------------
// Pass 1: per-(receiver, channel) normalization sums.
// ---------------------------------------------------------------------------
__global__ __launch_bounds__(256) void edge_sums_kernel(
        const float* __restrict__ dist, const int* __restrict__ recv,
        const float* __restrict__ w1, const float* __restrict__ b1,
        const float* __restrict__ w2, const float* __restrict__ b2,
        float* __restrict__ sums, int Ne) {
    __shared__ float sW1[HID], sB1[HID], sW2[HID * KM], sB2[KM];
    int tid = threadIdx.x;
    if (tid < HID) { sW1[tid] = w1[tid]; sB1[tid] = b1[tid]; }
    if (tid < KM)  { sB2[tid] = b2[tid]; }
    for (int i = tid; i < HID * KM; i += 256) sW2[i] = w2[i];
    __syncthreads();

    int gwave = (blockIdx.x * blockDim.x + threadIdx.x) >> 5;
    int lane  = threadIdx.x & 31;
    int ntile = (Ne + 15) >> 4;
    if (gwave >= ntile) return;               // wave-uniform
    int tileBase = gwave * 16;

    v8f acc[2];
    mlp_tile_wmma(tileBase, Ne, dist, sW1, sB1, sW2, sB2, lane, acc);

    // indicator one-hot counts: lanes 0-15, one edge each
    if (lane < 16) {
        int e = tileBase + lane;
        if (e < Ne) {
            int bin = indicator_bin(dist[e]);
            if (bin >= 0) atomicAdd(&sums[recv[e] * DIM + bin], 1.0f);
        }
    }

    // mlp sums
    int mbase = (lane >> 4) * 8;
    int coln  = lane & 15;
#pragma unroll
    for (int c = 0; c < 2; ++c) {
#pragma unroll
        for (int r = 0; r < 8; ++r) {
            int e = tileBase + mbase + r;
            if (e < Ne) {
                float v = fmaxf(acc[c][r], 0.0f);
                atomicAdd(&sums[recv[e] * DIM + KI + c * 16 + coln], v);
            }
        }
    }
}

// ---------------------------------------------------------------------------
// Pass 2: normalized weights * power-difference, scattered into gathered[N,64]
// ---------------------------------------------------------------------------
__global__ __launch_bounds__(256) void edge_gather_kernel(
        const float* __restrict__ nodes, const float* __restrict__ dist,
        const float* __restrict__ pad,
        const int* __restrict__ recv, const int* __restrict__ send,
        const float* __restrict__ w1, const float* __restrict__ b1,
        const float* __restrict__ w2, const float* __restrict__ b2,
        const float* __restrict__ pd_a, const float* __restrict__ pd_b,
        const float* __restrict__ sums, float* __restrict__ gathered, int Ne) {
    __shared__ float sW1[HID], sB1[HID], sW2[HID * KM], sB2[KM];
    int tid = threadIdx.x;
    if (tid < HID) { sW1[tid] = w1[tid]; sB1[tid] = b1[tid]; }
    if (tid < KM)  { sB2[tid] = b2[tid]; }
    for (int i = tid; i < HID * KM; i += 256) sW2[i] = w2[i];
    __syncthreads();

    int gwave = (blockIdx.x * blockDim.x + threadIdx.x) >> 5;
    int lane  = threadIdx.x & 31;
    int ntile = (Ne + 15) >> 4;
    if (gwave >= ntile) return;               // wave-uniform
    int tileBase = gwave * 16;

    float a  = fminf(fmaxf(pd_a[0], 0.0f), 1.0f);
    float oa = 1.0f - a;
    float bp = fabsf(pd_b[0]);

    v8f acc[2];
    mlp_tile_wmma(tileBase, Ne, dist, sW1, sB1, sW2, sB2, lane, acc);

    // indicator contribution (one nonzero channel per edge)
    if (lane < 16) {
        int e = tileBase + lane;
        if (e < Ne) {
            int bin = indicator_bin(dist[e]);
            if (bin >= 0) {
                int rv = recv[e], sd = send[e];
                float wv = pad[e] / (sums[rv * DIM + bin] + 1e-5f);
                float x  = fabsf(a * nodes[rv * DIM + bin] -
                                 oa * nodes[sd * DIM + bin]);
                atomicAdd(&gathered[rv * DIM + bin], wv * powf(x, bp));
            }
        }
    }

    // mlp-weight contributions (channels 32..63)
    int mbase = (lane >> 4) * 8;
    int coln  = lane & 15;
#pragma unroll
    for (int c = 0; c < 2; ++c) {
#pragma unroll
        for (int r = 0; r < 8; ++r) {
            int e = tileBase + mbase + r;
            if (e < Ne) {
                int k  = KI + c * 16 + coln;
                int rv = recv[e], sd = send[e];
                float v  = fmaxf(acc[c][r], 0.0f);
                float wv = v * pad[e] / (sums[rv * DIM + k] + 1e-5f);
                float x  = fabsf(a * nodes[rv * DIM + k] -
                                 oa * nodes[sd * DIM + k]);
                atomicAdd(&gathered[rv * DIM + k], wv * powf(x, bp));
            }
        }
    }
}

// ---------------------------------------------------------------------------
// Pass 3: out = relu(nodes @ gsw + gathered @ ggw + gb), all via f32 WMMA.
// Both 64x64 B matrices staged in LDS once per block (32 KB of 320 KB WGP LDS);
// A fragments hoisted to registers and reused across all 4 column tiles.
// ---------------------------------------------------------------------------
__global__ __launch_bounds__(256) void out_gemm_kernel(
        const float* __restrict__ nodes, const float* __restrict__ gathered,
        const float* __restrict__ gsw, const float* __restrict__ ggw,
        const float* __restrict__ gb, float* __restrict__ out, int Nn) {
    __shared__ float sGsw[DIM * DIM];   // 16 KB
    __shared__ float sGgw[DIM * DIM];   // 16 KB
    __shared__ float sGb[DIM];
    int tid = threadIdx.x;
    for (int i = tid; i < DIM * DIM; i += 256) {
        sGsw[i] = gsw[i];
        sGgw[i] = ggw[i];
    }
    if (tid < DIM) sGb[tid] = gb[tid];
    __syncthreads();

    int gwave = (blockIdx.x * blockDim.x + threadIdx.x) >> 5;
    int lane  = threadIdx.x & 31;
    int ntile = (Nn + 15) >> 4;
    if (gwave >= ntile) return;               // wave-uniform
    int tileBase = gwave * 16;

    int arow  = min(tileBase + (lane & 15), Nn - 1);
    int khalf = (lane >> 4) * 2;
    int coln  = lane & 15;
    int mbase = (lane >> 4) * 8;

    // hoisted A fragments for both sources (reused across 4 column tiles)
    v2f aS[16], aG[16];
#pragma unroll
    for (int ks = 0; ks < 16; ++ks) {
        int k0 = ks * 4 + khalf;
        aS[ks].x = nodes[arow * DIM + k0];
        aS[ks].y = nodes[arow * DIM + k0 + 1];
        aG[ks].x = gathered[arow * DIM + k0];
        aG[ks].y = gathered[arow * DIM + k0 + 1];
    }

#pragma unroll
    for (int c = 0; c < 4; ++c) {
        float bias = sGb[c * 16 + coln];
        v8f acc;
#pragma unroll
        for (int r = 0; r < 8; ++r) acc[r] = bias;

#pragma unroll
        for (int ks = 0; ks < 16; ++ks) {      // nodes @ gamma_self_w
            int k0 = ks * 4 + khalf;
            v2f bv;
            bv.x = sGsw[k0       * DIM + c * 16 + coln];
            bv.y = sGsw[(k0 + 1) * DIM + c * 16 + coln];
            acc = wmma_f32_16x16x4(aS[ks], bv, acc);
        }
#pragma unroll
        for (int ks = 0; ks < 16; ++ks) {      // gathered @ gamma_gathered_w
            int k0 = ks * 4 + khalf;
            v2f bv;
            bv.x = sGgw[k0       * DIM + c * 16 + coln];
            bv.y = sGgw[(k0 + 1) * DIM + c * 16 + coln];
            acc = wmma_f32_16x16x4(aG[ks], bv, acc);
        }

#pragma unroll
        for (int r = 0; r < 8; ++r) {
            int row = tileBase + mbase + r;
            if (row < Nn)
                out[row * DIM + c * 16 + coln] = fmaxf(acc[r], 0.0f);
        }
    }
}

// ---------------------------------------------------------------------------
extern "C" void kernel_launch(void* const* d_in, const int* in_sizes, int n_in,
                              void* d_out, int out_size, void* d_ws, size_t ws_size,
                              hipStream_t stream) {
    const float* nodes    = (const float*)d_in[0];
    const float* distance = (const float*)d_in[1];
    const float* padding  = (const float*)d_in[2];
    const float* mlp_w1   = (const float*)d_in[3];
    const float* mlp_b1   = (const float*)d_in[4];
    const float* mlp_w2   = (const float*)d_in[5];
    const float* mlp_b2   = (const float*)d_in[6];
    const float* pd_a     = (const float*)d_in[7];
    const float* pd_b     = (const float*)d_in[8];
    const float* gsw      = (const float*)d_in[9];
    const float* ggw      = (const float*)d_in[10];
    const float* gb       = (const float*)d_in[11];
    const int*   recv     = (const int*)d_in[12];
    const int*   send     = (const int*)d_in[13];
    float*       out      = (float*)d_out;

    const int Nn = in_sizes[0] / DIM;   // 50000 nodes
    const int Ne = in_sizes[1];         // 800000 edges

    float* sums     = (float*)d_ws;             // [Nn, 64]
    float* gathered = sums + (size_t)Nn * DIM;  // [Nn, 64]

    // 1) zero workspace
    int zn = 2 * Nn * DIM;
    zero_ws_kernel<<<(zn + 255) / 256, 256, 0, stream>>>(sums, zn);

    // 2) per-(receiver, channel) normalization sums
    int etiles   = (Ne + 15) / 16;
    int ethreads = etiles * 32;
    edge_sums_kernel<<<(ethreads + 255) / 256, 256, 0, stream>>>(
        distance, recv, mlp_w1, mlp_b1, mlp_w2, mlp_b2, sums, Ne);

    // 3) normalized weighted power-difference scatter
    edge_gather_kernel<<<(ethreads + 255) / 256, 256, 0, stream>>>(
        nodes, distance, padding, recv, send,
        mlp_w1, mlp_b1, mlp_w2, mlp_b2, pd_a, pd_b, sums, gathered, Ne);

    // 4) fused output GEMMs + bias + relu
    int ntiles   = (Nn + 15) / 16;
    int nthreads = ntiles * 32;
    out_gemm_kernel<<<(nthreads + 255) / 256, 256, 0, stream>>>(
        nodes, gathered, gsw, ggw, gb, out, Nn);
}